// SpectralConv1D_30717606101418
// MI455X (gfx1250) — compile-verified
//
#include <hip/hip_runtime.h>
#include <hip/hip_bf16.h>

// SpectralConv1D (FNO layer): truncated-DFT reformulation for MI455X (gfx1250).
// B=16, NX=8192, C=128, K1=17 retained modes.
// Stage 0 : build cos/sin basis (48x8192 row-major + 8192x48 transposed).
// Stage 1 : partial fx = basis @ x  (fp32 WMMA 16x16x4; K=8192 split into 4
//           slabs x 8 waves for occupancy; deterministic LDS tree-reduce).
// Stage 1b: fixed-order reduction of the 4 slab partials.
// Stage 2 : per-mode complex channel mix + irfft scaling into real matrix G.
// Stage 3 : y = ibasis @ G  (fp32 WMMA 16x16x4, K=48, A reused across ctiles).

#define B_    16
#define NX_   8192
#define C_    128
#define K1_   17
#define ROWS_ 48                  // 2*K1 = 34 padded to 3 WMMA row-tiles
#define SLABS_ 4                  // split-K factor for stage 1
#define TWO_PI_F 6.2831853071795864769f

typedef __attribute__((ext_vector_type(2))) float v2f;
typedef __attribute__((ext_vector_type(8))) float v8f;

// ---------------- Stage 0: basis generation ----------------
__global__ __launch_bounds__(256)
void fno_basis(float* __restrict__ fbasis, float* __restrict__ ibasisT) {
  int n = blockIdx.x * blockDim.x + threadIdx.x;
  if (n >= NX_) return;
  #pragma unroll 1
  for (int k = 0; k < K1_; ++k) {
    int ph = (k * n) & (NX_ - 1);               // exact phase mod NX
    float ang = (float)ph * (TWO_PI_F / (float)NX_);
    float s, c;
    sincosf(ang, &s, &c);
    fbasis[(size_t)(2*k+0)*NX_ + n] = c;
    fbasis[(size_t)(2*k+1)*NX_ + n] = s;
    ibasisT[(size_t)n*ROWS_ + 2*k+0] = c;
    ibasisT[(size_t)n*ROWS_ + 2*k+1] = s;
  }
  for (int r = 2*K1_; r < ROWS_; ++r) {         // zero pad rows (ws is poisoned)
    fbasis[(size_t)r*NX_ + n] = 0.0f;
    ibasisT[(size_t)n*ROWS_ + r] = 0.0f;
  }
}

// ---------------- Stage 1: forward truncated DFT (partials) ----------------
// grid: B*8*SLABS blocks (b, channel-tile, K-slab); 256 threads = 8 waves.
// Each wave owns a 256-position K-chunk and 3 row-tiles. Block reduces its
// 8 wave-partials in LDS (fixed order) and emits one partial 48x16 tile.
__global__ __launch_bounds__(256)
void fno_forward(const float* __restrict__ x, const float* __restrict__ fbasis,
                 float* __restrict__ pbuf) {
  __shared__ float red[8 * 768];                // 24 KB
  const int bi   = blockIdx.x;                  // b*32 + ct*4 + slab
  const int b    = bi >> 5;
  const int ct   = (bi >> 2) & 7;
  const int slab = bi & 3;
  const int c0   = ct * 16;
  const int wave = threadIdx.x >> 5;
  const int lane = threadIdx.x & 31;
  const int half = lane >> 4;                   // K-subgroup select
  const int l    = lane & 15;

  const float* xb = x + (size_t)b * NX_ * C_;
  v8f acc0 = {}, acc1 = {}, acc2 = {};
  const int nbase = slab * (NX_ / SLABS_) + wave * (NX_ / SLABS_ / 8); // 256/wave

  #pragma unroll 4
  for (int g = 0; g < (NX_ / SLABS_ / 8) / 4; ++g) {   // 64 K-groups
    const int n0 = nbase + g * 4 + 2 * half;
    // B tile (4x16 of x): lane l holds channel c0+l, K pair (n0, n0+1)
    v2f bb;
    bb.x = xb[(size_t)(n0    ) * C_ + c0 + l];
    bb.y = xb[(size_t)(n0 + 1) * C_ + c0 + l];
    // A tiles (16x4 of basis rows): contiguous 8B loads, L2-resident
    v2f a0 = *(const v2f*)(fbasis + (size_t)( 0 + l) * NX_ + n0);
    v2f a1 = *(const v2f*)(fbasis + (size_t)(16 + l) * NX_ + n0);
    v2f a2 = *(const v2f*)(fbasis + (size_t)(32 + l) * NX_ + n0);
    acc0 = __builtin_amdgcn_wmma_f32_16x16x4_f32(false, a0, false, bb, (short)0, acc0, false, false);
    acc1 = __builtin_amdgcn_wmma_f32_16x16x4_f32(false, a1, false, bb, (short)0, acc1, false, false);
    acc2 = __builtin_amdgcn_wmma_f32_16x16x4_f32(false, a2, false, bb, (short)0, acc2, false, false);
  }

  float* wr = red + wave * 768;
  #pragma unroll
  for (int v = 0; v < 8; ++v) {
    wr[0 * 256 + v * 32 + lane] = acc0[v];
    wr[1 * 256 + v * 32 + lane] = acc1[v];
    wr[2 * 256 + v * 32 + lane] = acc2[v];
  }
  __syncthreads();

  // fixed-order cross-wave reduction -> one partial tile per block
  const int tid = threadIdx.x;
  #pragma unroll
  for (int r = 0; r < 3; ++r) {
    float s = 0.0f;
    #pragma unroll
    for (int w = 0; w < 8; ++w) s += red[w * 768 + r * 256 + tid];
    pbuf[(size_t)bi * 768 + r * 256 + tid] = s;
  }
}

// ---------------- Stage 1b: fixed-order slab reduction ----------------
__global__ __launch_bounds__(256)
void fno_reduce(const float* __restrict__ pbuf, float* __restrict__ fxsum) {
  const int bt  = blockIdx.x;                   // b*8 + ct
  const int b   = bt >> 3;
  const int c0  = (bt & 7) * 16;
  const int tid = threadIdx.x;
  const int v     = tid >> 5;                   // C/D layout decode
  const int rl    = tid & 31;
  const int rhalf = rl >> 4;
  const int rcol  = rl & 15;
  #pragma unroll
  for (int r = 0; r < 3; ++r) {
    float s = 0.0f;
    #pragma unroll
    for (int slab = 0; slab < SLABS_; ++slab)
      s += pbuf[(size_t)(bt * SLABS_ + slab) * 768 + r * 256 + tid];
    const int grow = r * 16 + v + 8 * rhalf;    // M = vgpr + 8*half
    fxsum[((size_t)b * ROWS_ + grow) * C_ + c0 + rcol] = s;
  }
}

// ---------------- Stage 2: per-mode complex channel mix ----------------
// fxsum rows: 2k = Σ x·cos (=Re fx), 2k+1 = Σ x·sin (Im fx = -that).
// G[2k] = a_k*Re(W_k fx_k); G[2k+1] = -a_k*Im(W_k fx_k); a_0=1/NX, a_k=2/NX.
__global__ __launch_bounds__(128)
void fno_mix(const float* __restrict__ w_real, const float* __restrict__ w_imag,
             const float* __restrict__ fxsum, float* __restrict__ G) {
  const int b = blockIdx.x / 24;
  const int k = blockIdx.x % 24;
  const int i = threadIdx.x;
  if (k >= K1_) {                               // zero pad rows 34..47
    G[((size_t)b * ROWS_ + 2 * k + 0) * C_ + i] = 0.0f;
    G[((size_t)b * ROWS_ + 2 * k + 1) * C_ + i] = 0.0f;
    return;
  }
  const float* fr  = fxsum  + ((size_t)b * ROWS_ + 2 * k + 0) * C_;
  const float* fs  = fxsum  + ((size_t)b * ROWS_ + 2 * k + 1) * C_;
  const float* wrk = w_real + (size_t)k * C_ * C_ + (size_t)i * C_;
  const float* wik = w_imag + (size_t)k * C_ * C_ + (size_t)i * C_;
  float ar = 0.0f, ai = 0.0f;
  #pragma unroll 4
  for (int j = 0; j < C_; ++j) {
    const float xr = fr[j];
    const float xi = -fs[j];
    const float wr = wrk[j], wi = wik[j];
    ar = fmaf(wr, xr, fmaf(-wi, xi, ar));
    ai = fmaf(wr, xi, fmaf( wi, xr, ai));
  }
  const float a = (k == 0) ? (1.0f / NX_) : (2.0f / NX_);
  G[((size_t)b * ROWS_ + 2 * k + 0) * C_ + i] =  a * ar;
  G[((size_t)b * ROWS_ + 2 * k + 1) * C_ + i] = -a * ai;
}

// ---------------- Stage 3: inverse truncated DFT ----------------
// 1024 blocks x 8 waves; wave owns one 16-position n-tile of one batch,
// iterates 8 channel tiles reusing preloaded A (12 v2f regs), K=48.
__global__ __launch_bounds__(256)
void fno_inverse(const float* __restrict__ ibasisT, const float* __restrict__ G,
                 float* __restrict__ y) {
  const int wave = threadIdx.x >> 5;
  const int lane = threadIdx.x & 31;
  const int half = lane >> 4;
  const int l    = lane & 15;
  const int gw   = blockIdx.x * 8 + wave;
  const int b    = gw >> 9;                     // /512 n-tiles per batch
  const int n0   = (gw & 511) * 16;
  const float* Gb = G + (size_t)b * ROWS_ * C_;

  v2f aa[12];
  #pragma unroll
  for (int g = 0; g < 12; ++g)
    aa[g] = *(const v2f*)(ibasisT + (size_t)(n0 + l) * ROWS_ + g * 4 + 2 * half);

  #pragma unroll 1
  for (int ct = 0; ct < 8; ++ct) {
    const int c0 = ct * 16;
    v8f acc = {};
    #pragma unroll
    for (int g = 0; g < 12; ++g) {
      const int k0 = g * 4 + 2 * half;
      v2f bb;
      bb.x = Gb[(size_t)(k0    ) * C_ + c0 + l];
      bb.y = Gb[(size_t)(k0 + 1) * C_ + c0 + l];
      acc = __builtin_amdgcn_wmma_f32_16x16x4_f32(false, aa[g], false, bb, (short)0, acc, false, false);
    }
    #pragma unroll
    for (int v = 0; v < 8; ++v)
      y[((size_t)b * NX_ + n0 + v + 8 * half) * C_ + c0 + l] = acc[v];
  }
}

extern "C" void kernel_launch(void* const* d_in, const int* in_sizes, int n_in,
                              void* d_out, int out_size, void* d_ws, size_t ws_size,
                              hipStream_t stream) {
  const float* x      = (const float*)d_in[0];
  const float* w_real = (const float*)d_in[1];
  const float* w_imag = (const float*)d_in[2];
  float* out = (float*)d_out;

  float* fbasis  = (float*)d_ws;                          // 48*8192
  float* ibasisT = fbasis  + (size_t)ROWS_ * NX_;         // 8192*48
  float* fxsum   = ibasisT + (size_t)NX_ * ROWS_;         // B*48*C
  float* G       = fxsum   + (size_t)B_ * ROWS_ * C_;     // B*48*C
  float* pbuf    = G       + (size_t)B_ * ROWS_ * C_;     // B*8*SLABS*768
  (void)in_sizes; (void)n_in; (void)out_size; (void)ws_size;

  fno_basis  <<<NX_ / 256,          256, 0, stream>>>(fbasis, ibasisT);
  fno_forward<<<B_ * 8 * SLABS_,    256, 0, stream>>>(x, fbasis, pbuf);
  fno_reduce <<<B_ * 8,             256, 0, stream>>>(pbuf, fxsum);
  fno_mix    <<<B_ * 24,            128, 0, stream>>>(w_real, w_imag, fxsum, G);
  fno_inverse<<<(B_ * 512) / 8,     256, 0, stream>>>(ibasisT, G, out);
}